// GCNLayer_23596550324600
// MI455X (gfx1250) — compile-verified
//
#include <hip/hip_runtime.h>
#include <math.h>

// GCN layer: h = tanh( (segment_sum(feat[src]*norm[src], dst) * norm) @ W^T + b )
// N=50000 nodes, E=800000 edges, F=128 features (in==out).
//
// Kernel 1: zero agg (= d_out, reused in place)
// Kernel 2: edge scatter, 1 wave per edge, 4 floats/lane, L2 f32 atomics
// Kernel 3: fp32 WMMA GEMM (16x16x4_f32) + bias + tanh, in-place on d_out

typedef __attribute__((ext_vector_type(2))) float v2f;
typedef __attribute__((ext_vector_type(8))) float v8f;

#define N_NODES 50000
#define FEATS   128
#define LDS_STRIDE 132  // 128 + 4 pad: conflict-free across half-wave K offsets

__global__ void gcn_zero_kernel(float4* __restrict__ p, int n4) {
    int i = blockIdx.x * blockDim.x + threadIdx.x;
    if (i < n4) p[i] = make_float4(0.f, 0.f, 0.f, 0.f);
}

// One wave (32 lanes) per edge; each lane handles a float4 slice of the
// 128-wide feature row. Gather feat[src]*norm[src], atomic-add into agg[dst].
__global__ void gcn_edge_scatter(const float* __restrict__ feats,
                                 const float* __restrict__ norm,
                                 const int*   __restrict__ src,
                                 const int*   __restrict__ dst,
                                 float*       __restrict__ agg,
                                 int n_edges) {
    int e = blockIdx.x * 8 + (threadIdx.x >> 5);   // 8 waves per block
    if (e >= n_edges) return;
    int lane = threadIdx.x & 31;
    int s = src[e];
    int d = dst[e];
    float ns = norm[s];
    float4 v = ((const float4*)(feats + (size_t)s * FEATS))[lane];
    float* out = agg + (size_t)d * FEATS + (size_t)lane * 4;
    atomicAdd(out + 0, v.x * ns);
    atomicAdd(out + 1, v.y * ns);
    atomicAdd(out + 2, v.z * ns);
    atomicAdd(out + 3, v.w * ns);
}

// Block = 256 threads = 8 wave32; handles 16 rows of the output.
// Wave w computes the 16x16 tile for output columns [16w, 16w+16).
// A (agg*norm) tile 16x128 staged in LDS once, shared by all 8 waves.
// B fragment comes from W (row-major [128][128]): B[k][n] = W[n][k], read
// directly as rows of W (no transpose needed in registers).
//
// f32 WMMA fragment layout (16x16x4):
//   A 16x4: lane (half*16 + r): VGPR0 = A[r][k0+2*half], VGPR1 = A[r][k0+2*half+1]
//   B 4x16: lane (half*16 + r): VGPR0 = B[k0+2*half][r], VGPR1 = B[k0+2*half+1][r]
//   C/D 16x16: VGPR v, lane (half*16 + r) -> row (v + 8*half), col r
__global__ void gcn_gemm_bias_tanh(float* __restrict__ agg,      // in-place [N,128]
                                   const float* __restrict__ norm,
                                   const float* __restrict__ W,   // [128,128] row-major
                                   const float* __restrict__ bias) {
    __shared__ float As[16 * LDS_STRIDE];

    int row0 = blockIdx.x * 16;

    // Cooperative load of the 16x128 A tile, folding in dst-side norm.
    for (int i = threadIdx.x; i < 16 * FEATS; i += 256) {
        int r = i >> 7;
        int k = i & 127;
        int n = row0 + r;
        As[r * LDS_STRIDE + k] = agg[(size_t)n * FEATS + k] * norm[n];
    }
    __syncthreads();

    int wave = threadIdx.x >> 5;      // 0..7 -> column tile
    int lane = threadIdx.x & 31;
    int half = lane >> 4;             // 0 or 1
    int r    = lane & 15;
    int col  = wave * 16 + r;

    const float* wrow = W + (size_t)col * FEATS;   // W[col][k] == B[k][col]

    v8f c = {};
    #pragma unroll
    for (int k0 = 0; k0 < FEATS; k0 += 4) {
        int kk = k0 + 2 * half;
        v2f a, b;
        a.x = As[r * LDS_STRIDE + kk];
        a.y = As[r * LDS_STRIDE + kk + 1];
        b.x = wrow[kk];
        b.y = wrow[kk + 1];
        c = __builtin_amdgcn_wmma_f32_16x16x4_f32(
                /*neg_a=*/false, a, /*neg_b=*/false, b,
                /*c_mod=*/(short)0, c, /*reuse_a=*/false, /*reuse_b=*/false);
    }

    float bb = bias[col];
    #pragma unroll
    for (int v = 0; v < 8; v++) {
        int m = row0 + v + 8 * half;
        agg[(size_t)m * FEATS + col] = tanhf(c[v] + bb);
    }
}

extern "C" void kernel_launch(void* const* d_in, const int* in_sizes, int n_in,
                              void* d_out, int out_size, void* d_ws, size_t ws_size,
                              hipStream_t stream) {
    const float* feats = (const float*)d_in[0];  // [N,128]
    const float* norm  = (const float*)d_in[1];  // [N,1]
    const float* W     = (const float*)d_in[2];  // [128,128]
    const float* bias  = (const float*)d_in[3];  // [128]
    const int*   src   = (const int*)d_in[4];    // [E]
    const int*   dst   = (const int*)d_in[5];    // [E]
    int n_edges = in_sizes[4];

    float* agg = (float*)d_out;                  // reused: agg then final h

    int n4 = (N_NODES * FEATS) / 4;              // 1.6M float4
    gcn_zero_kernel<<<(n4 + 255) / 256, 256, 0, stream>>>((float4*)agg, n4);

    gcn_edge_scatter<<<(n_edges + 7) / 8, 256, 0, stream>>>(
        feats, norm, src, dst, agg, n_edges);

    gcn_gemm_bias_tanh<<<N_NODES / 16, 256, 0, stream>>>(agg, norm, W, bias);
}